// GenomeNet_86552180949490
// MI455X (gfx1250) — compile-verified
//
#include <hip/hip_runtime.h>
#include <hip/hip_bf16.h>
#include <cstdint>
#include <math.h>

// Problem constants (match reference)
#define B_     4096
#define NIN_   1024
#define L_     8
#define W_     1024
#define NOUT_  1024
#define D_     16

#define NPB    8            // nodes per block
#define TPB    256          // 8 wave32 per block
#define HALFB  (B_ / 2)     // float2 columns per activation row

// CDNA5 async global->LDS path (gfx1250). Guarded so compilation never breaks.
#if __has_builtin(__builtin_amdgcn_global_load_async_to_lds_b32) && \
    __has_builtin(__builtin_amdgcn_s_wait_asynccnt)
#define USE_ASYNC_LDS 1
#else
#define USE_ASYNC_LDS 0
#endif

#if USE_ASYNC_LDS
// Builtin prototype (per hipcc diagnostic) takes int* in AS(1) / AS(3).
#define AS1_INT(p) ((__attribute__((address_space(1))) int*)(uintptr_t)(p))
#define AS3_INT(p) ((__attribute__((address_space(3))) int*)(int*)(p))
#endif

// ---------------------------------------------------------------------------
// Gather layer: vout[w, b] = act( sum_d wgt[w,d] * vin[src[w,d], b] )
// Activations live transposed as [feat, B] so gathers are coalesced across
// the batch (lanes). Indices/weights are wave-uniform -> staged in LDS once
// per block via the gfx1250 async-to-LDS DMA path.
// ---------------------------------------------------------------------------
__global__ __launch_bounds__(TPB) void gather_layer_kernel(
    const float* __restrict__ vin,   // [feat_prev, B]
    const int*   __restrict__ src,   // [Wn, D]
    const float* __restrict__ wgt,   // [Wn, D]
    float*       __restrict__ vout,  // [Wn, B]
    int apply_tanh)
{
    __shared__ int   s_idx[NPB * D_];
    __shared__ float s_w  [NPB * D_];

    const int tid  = threadIdx.x;
    const int gofs = blockIdx.y * (NPB * D_);

#if USE_ASYNC_LDS
    if (tid < NPB * D_) {
        __builtin_amdgcn_global_load_async_to_lds_b32(
            AS1_INT(src + gofs + tid), AS3_INT(&s_idx[tid]), 0, 0);
    } else {
        __builtin_amdgcn_global_load_async_to_lds_b32(
            AS1_INT(wgt + gofs + (tid - NPB * D_)),
            AS3_INT(&s_w[tid - NPB * D_]), 0, 0);
    }
    __builtin_amdgcn_s_wait_asynccnt(0);
#else
    if (tid < NPB * D_) s_idx[tid] = src[gofs + tid];
    else                s_w[tid - NPB * D_] = wgt[gofs + (tid - NPB * D_)];
#endif
    __syncthreads();

    const int bt = blockIdx.x * TPB + tid;             // float2 column, 0..HALFB-1
    const float2* __restrict__ vin2  = (const float2*)vin;
    float2*       __restrict__ vout2 = (float2*)vout;

    for (int n = 0; n < NPB; ++n) {
        float ax = 0.f, ay = 0.f;
#pragma unroll
        for (int d = 0; d < D_; ++d) {
            const int   j  = s_idx[n * D_ + d];        // wave-uniform ds broadcast
            const float wv = s_w [n * D_ + d];
            const float2 v = vin2[(size_t)j * HALFB + bt];  // coalesced gather
            ax = fmaf(wv, v.x, ax);
            ay = fmaf(wv, v.y, ay);
        }
        if (apply_tanh) { ax = tanhf(ax); ay = tanhf(ay); }
        const int w = blockIdx.y * NPB + n;
        vout2[(size_t)w * HALFB + bt] = make_float2(ax, ay);
    }
}

// ---------------------------------------------------------------------------
// LDS-tiled transpose: out[c][r] = in[r][c].  rows/cols multiples of 32.
// grid = (cols/32, rows/32), block = (32, 8)
// ---------------------------------------------------------------------------
__global__ __launch_bounds__(256) void transpose_kernel(
    const float* __restrict__ in, float* __restrict__ out, int rows, int cols)
{
    __shared__ float tile[32][33];
    const int x = blockIdx.x * 32 + threadIdx.x;
    const int y = blockIdx.y * 32 + threadIdx.y;
#pragma unroll
    for (int j = 0; j < 32; j += 8)
        tile[threadIdx.y + j][threadIdx.x] = in[(size_t)(y + j) * cols + x];
    __syncthreads();
    const int ox = blockIdx.y * 32 + threadIdx.x;
#pragma unroll
    for (int j = 0; j < 32; j += 8)
        out[(size_t)(blockIdx.x * 32 + threadIdx.y + j) * rows + ox] =
            tile[threadIdx.x][threadIdx.y + j];
}

// ---------------------------------------------------------------------------
extern "C" void kernel_launch(void* const* d_in, const int* in_sizes, int n_in,
                              void* d_out, int out_size, void* d_ws, size_t ws_size,
                              hipStream_t stream)
{
    (void)in_sizes; (void)n_in; (void)out_size;

    const float* x    = (const float*)d_in[0];   // [B, N_IN]
    const int*   srcH = (const int*)  d_in[1];   // [L, W, D]
    const float* wH   = (const float*)d_in[2];   // [L, W, D]
    const int*   srcO = (const int*)  d_in[3];   // [N_OUT, D]
    const float* wO   = (const float*)d_in[4];   // [N_OUT, D]
    float* out = (float*)d_out;                  // [B, N_OUT]

    const size_t BUF_ELEMS = (size_t)W_ * B_;    // one activation plane (16 MB)
    float* t0 = (float*)d_ws;
    float* t1;
    bool via_dout = false;
    if (ws_size >= 2 * BUF_ELEMS * sizeof(float)) {
        t1 = t0 + BUF_ELEMS;                     // pure workspace ping-pong
    } else {
        t1 = out;                                // borrow d_out as second plane
        via_dout = true;
    }

    const dim3 tb(32, 8);

    // x [B, N_IN] -> t0 [N_IN, B]
    transpose_kernel<<<dim3(NIN_ / 32, B_ / 32), tb, 0, stream>>>(x, t0, B_, NIN_);

    // 8 hidden gather layers, ping-pong t0 <-> t1
    const dim3 gb(HALFB / TPB, W_ / NPB);        // (8, 128)
    float* cur = t0;
    float* nxt = t1;
    for (int l = 0; l < L_; ++l) {
        gather_layer_kernel<<<gb, TPB, 0, stream>>>(
            cur, srcH + (size_t)l * W_ * D_, wH + (size_t)l * W_ * D_, nxt, 1);
        float* tmp = cur; cur = nxt; nxt = tmp;
    }
    // after an even number of layers: cur == t0, nxt == t1

    // output layer (identity activation): cur -> nxt holds out_T [N_OUT, B]
    gather_layer_kernel<<<dim3(HALFB / TPB, NOUT_ / NPB), TPB, 0, stream>>>(
        cur, srcO, wO, nxt, 0);

    // out_T [N_OUT, B] -> d_out [B, N_OUT]
    if (!via_dout) {
        transpose_kernel<<<dim3(B_ / 32, NOUT_ / 32), tb, 0, stream>>>(nxt, out, NOUT_, B_);
    } else {
        // nxt aliases d_out: transpose into t0, then device-to-device copy
        transpose_kernel<<<dim3(B_ / 32, NOUT_ / 32), tb, 0, stream>>>(nxt, t0, NOUT_, B_);
        (void)hipMemcpyAsync(out, t0, BUF_ELEMS * sizeof(float),
                             hipMemcpyDeviceToDevice, stream);
    }
}